// QLSTM_85804856640302
// MI455X (gfx1250) — compile-verified
//
#include <hip/hip_runtime.h>
#include <hip/hip_bf16.h>
#include <math.h>

typedef __attribute__((ext_vector_type(2))) float v2f;
typedef __attribute__((ext_vector_type(8))) float v8f;

static __device__ __forceinline__ v8f wmma_f32_16x16x4(v2f a, v2f b, v8f c) {
  // D = A(16x4) * B(4x16) + C(16x16), fp32, one wave
  return __builtin_amdgcn_wmma_f32_16x16x4_f32(false, a, false, b, (short)0, c, false, false);
}

static __device__ __forceinline__ float fast_sigmoid(float x) {
  x = fminf(fmaxf(x, -30.f), 30.f);
  return __builtin_amdgcn_rcpf(1.f + __expf(-x));   // v_exp_f32 + v_rcp_f32
}
static __device__ __forceinline__ float fast_tanh(float x) {
  x = fminf(fmaxf(x, -15.f), 15.f);
  float e = __expf(-2.f * x);
  return (1.f - e) * __builtin_amdgcn_rcpf(1.f + e);
}

#define B_SZ 256
#define S_SZ 256
#define F_SZ 64
#define H_SZ 128
#define HSTR 132   // padded LDS row stride for h (kills 16-way bank conflicts)

// ---------------------------------------------------------------------------
// Kernel 1: y0[(b*S+t)*4 + q] = sum_f x[b,t,f] * W_in[q][128+f] + b_in[q]
// GEMM M=65536, K=64, N=4 (padded to 16) via V_WMMA_F32_16X16X4_F32.
// One wave per 16-row tile; 4 waves per block. b_in folded into C init.
// ---------------------------------------------------------------------------
__global__ void __launch_bounds__(128)
qlstm_xproj(const float* __restrict__ x, const float* __restrict__ Win,
            const float* __restrict__ bin, float* __restrict__ y0) {
  const int lane = threadIdx.x & 31;
  const int wave = threadIdx.x >> 5;
  const int tile = blockIdx.x * 4 + wave;       // 0..4095
  const int n    = lane & 15;                   // N col (only n<4 meaningful)
  const int hi   = lane >> 4;                   // half-wave
  const int kh   = hi << 1;                     // K sub-offset: 0 or 2

  // Preload B chunks: B[kk][n] = Win[n][128+kk], zero-padded for n>=4.
  v2f bch[16];
#pragma unroll
  for (int j = 0; j < 16; ++j) {
    float b0 = 0.f, b1 = 0.f;
    if (n < 4) {
      int kk = j * 4 + kh;
      b0 = Win[n * 192 + 128 + kk];
      b1 = Win[n * 192 + 128 + kk + 1];
    }
    v2f bv = {b0, b1};
    bch[j] = bv;
  }

  const float bb = (n < 4) ? bin[n] : 0.f;      // C[m][n] = b_in[n] (splat)
  v8f acc = {bb, bb, bb, bb, bb, bb, bb, bb};

  const float* xr = x + (size_t)(tile * 16 + n) * F_SZ;  // A row per lane
#pragma unroll
  for (int j = 0; j < 16; ++j) {
    v2f a = {xr[j * 4 + kh], xr[j * 4 + kh + 1]};
    acc = wmma_f32_16x16x4(a, bch[j], acc);
  }

  if (n < 4) {
#pragma unroll
    for (int r = 0; r < 8; ++r) {
      int row = tile * 16 + r + (hi << 3);
      y0[(size_t)row * 4 + n] = acc[r];
    }
  }
}

// ---------------------------------------------------------------------------
// Kernel 2: recurrent scan. 16 blocks x 256 threads; block owns 16 batch rows.
// y0 slice (contiguous 64KB) staged into LDS once. Per step: (1) y-partials
// via WMMA over wave's K-slice of h @ W_in_h^T, (2) analytic 4-qubit
// product-state sim (half-angle, 1x v_cos_f32 per wire), (3) per-wave gate
// WMMA (C pre-loaded with b_out broadcast) + LSTM update with native
// exp/rcp activations; c kept in registers, h in LDS.
// ---------------------------------------------------------------------------
__global__ void __launch_bounds__(256)
qlstm_recur(const float* __restrict__ y0, const float* __restrict__ Win,
            const float* __restrict__ Wout, const float* __restrict__ bout,
            const float* __restrict__ thetas, float* __restrict__ out) {
  __shared__ float y0_lds[16 * S_SZ * 4];  // 64 KB: this block's y0 slice
  __shared__ float h_lds[16 * HSTR];
  __shared__ float part[8 * 16 * 4];       // per-wave partial y (16 batch x 4)
  __shared__ float z_lds[4 * 16 * 4];      // per-gate qlayer output

  const int tid   = threadIdx.x;
  const int lane  = tid & 31;
  const int wave  = tid >> 5;              // 0..7, owns H cols [wave*16, +16)
  const int n     = lane & 15;
  const int hi    = lane >> 4;
  const int kh    = hi << 1;
  const int bbase = blockIdx.x * 16;

  float* hs = out;
  float* hT = out + (size_t)B_SZ * S_SZ * H_SZ;
  float* cT = hT + (size_t)B_SZ * H_SZ;

  // Stage y0 tile: global range [bbase*S*4, +16*S*4) is contiguous.
  {
    const float4* src = (const float4*)(y0 + (size_t)bbase * S_SZ * 4);
    float4* dst = (float4*)y0_lds;
#pragma unroll
    for (int i = 0; i < 16; ++i)           // 4096 float4 / 256 threads
      dst[tid + i * 256] = src[tid + i * 256];
  }
  for (int i = tid; i < 16 * HSTR; i += 256) h_lds[i] = 0.f;

  // Preload W_in_h B-chunks for this wave's K range: B[kk][q]=Win[q][kk].
  v2f bh[4];
#pragma unroll
  for (int j = 0; j < 4; ++j) {
    float b0 = 0.f, b1 = 0.f;
    if (n < 4) {
      int kk = wave * 16 + j * 4 + kh;
      b0 = Win[n * 192 + kk];
      b1 = Win[n * 192 + kk + 1];
    }
    v2f bv = {b0, b1};
    bh[j] = bv;
  }
  // Preload W_out B slice: B[kk][j] = Wout[(wave*16+j)][kk]
  const int col = wave * 16 + n;
  v2f bw = {Wout[col * 4 + kh], Wout[col * 4 + kh + 1]};
  const float bo = bout[col];
  const v8f cbias = {bo, bo, bo, bo, bo, bo, bo, bo};  // C[m][n]=b_out[n]

  // Sim-thread constants (thread = (batch b, gate g) for tid<64)
  const int sb = tid >> 2, sg = tid & 3;
  float th[4];
#pragma unroll
  for (int w = 0; w < 4; ++w) th[w] = (tid < 64) ? thetas[sg * 4 + w] : 0.f;

  float c_state[8], h_keep[8];
#pragma unroll
  for (int r = 0; r < 8; ++r) { c_state[r] = 0.f; h_keep[r] = 0.f; }

  __syncthreads();

  for (int t = 0; t < S_SZ; ++t) {
    // ---- (1) partial y = h(16x128) @ W_in_h^T, this wave's 16-wide K slice
    v8f acc = {0.f, 0.f, 0.f, 0.f, 0.f, 0.f, 0.f, 0.f};
#pragma unroll
    for (int j = 0; j < 4; ++j) {
      int kk = wave * 16 + j * 4 + kh;
      v2f a = {h_lds[n * HSTR + kk], h_lds[n * HSTR + kk + 1]};
      acc = wmma_f32_16x16x4(a, bh[j], acc);
    }
    if (n < 4) {
#pragma unroll
      for (int r = 0; r < 8; ++r)
        part[wave * 64 + (r + (hi << 3)) * 4 + n] = acc[r];
    }
    __syncthreads();

    // ---- (2) 4-qubit sim: product state => probs via half-angle identity
    //      cos^2(a/2) = (1+cos a)/2 ; CNOT ring is a basis permutation.
    if (tid < 64) {
      float p0[4], p1[4];
#pragma unroll
      for (int w = 0; w < 4; ++w) {
        float yw = y0_lds[(sb * S_SZ + t) * 4 + w];
#pragma unroll
        for (int wv = 0; wv < 8; ++wv) yw += part[wv * 64 + sb * 4 + w];
        float cv = __cosf(yw + th[w]);          // one v_cos_f32 per wire
        p0[w] = 0.5f + 0.5f * cv;
        p1[w] = 0.5f - 0.5f * cv;
      }
      float z0 = 0.f, z1 = 0.f, z2 = 0.f, z3 = 0.f;
#pragma unroll
      for (int idx = 0; idx < 16; ++idx) {
        int a0 = (idx >> 3) & 1, a1 = (idx >> 2) & 1;
        int a2 = (idx >> 1) & 1, a3 = idx & 1;
        float p = (a0 ? p1[0] : p0[0]) * (a1 ? p1[1] : p0[1]) *
                  (a2 ? p1[2] : p0[2]) * (a3 ? p1[3] : p0[3]);
        int t1 = a1 ^ a0;      // CNOT(0,1)
        int t2 = a2 ^ t1;      // CNOT(1,2)
        int t3 = a3 ^ t2;      // CNOT(2,3)
        int t0 = a0 ^ t3;      // CNOT(3,0)
        z0 += t0 ? -p : p;  z1 += t1 ? -p : p;
        z2 += t2 ? -p : p;  z3 += t3 ? -p : p;
      }
      int base = (sg * 16 + sb) * 4;
      z_lds[base + 0] = z0;  z_lds[base + 1] = z1;
      z_lds[base + 2] = z2;  z_lds[base + 3] = z3;
    }
    __syncthreads();

    // ---- (3) gates: z_g(16x4) @ W_out^T slice + b_out, 1 WMMA/gate
    v8f ga[4];
#pragma unroll
    for (int g = 0; g < 4; ++g) {
      v2f a = {z_lds[(g * 16 + n) * 4 + kh], z_lds[(g * 16 + n) * 4 + kh + 1]};
      ga[g] = wmma_f32_16x16x4(a, bw, cbias);
    }
#pragma unroll
    for (int r = 0; r < 8; ++r) {
      float fv = fast_sigmoid(ga[0][r]);
      float iv = fast_sigmoid(ga[1][r]);
      float gv = fast_tanh(ga[2][r]);
      float ov = fast_sigmoid(ga[3][r]);
      float cs = fv * c_state[r] + iv * gv;
      c_state[r] = cs;
      float hv = ov * fast_tanh(cs);
      h_keep[r] = hv;
      int mrow = r + (hi << 3);
      h_lds[mrow * HSTR + col] = hv;
      hs[(((size_t)(bbase + mrow)) * S_SZ + t) * H_SZ + col] = hv;
    }
    __syncthreads();
  }

#pragma unroll
  for (int r = 0; r < 8; ++r) {
    int mrow = r + (hi << 3);
    hT[((size_t)(bbase + mrow)) * H_SZ + col] = h_keep[r];
    cT[((size_t)(bbase + mrow)) * H_SZ + col] = c_state[r];
  }
}

// ---------------------------------------------------------------------------
extern "C" void kernel_launch(void* const* d_in, const int* in_sizes, int n_in,
                              void* d_out, int out_size, void* d_ws, size_t ws_size,
                              hipStream_t stream) {
  const float* x      = (const float*)d_in[0];
  const float* W_in   = (const float*)d_in[1];
  const float* b_in   = (const float*)d_in[2];
  const float* W_out  = (const float*)d_in[3];
  const float* b_out  = (const float*)d_in[4];
  const float* thetas = (const float*)d_in[5];
  float* out = (float*)d_out;
  float* y0  = (float*)d_ws;   // (B*S, 4) = 1 MB of scratch

  qlstm_xproj<<<1024, 128, 0, stream>>>(x, W_in, b_in, y0);
  qlstm_recur<<<16, 256, 0, stream>>>(y0, W_in, W_out, b_out, thetas, out);
}